// PSM_50087908606629
// MI455X (gfx1250) — compile-verified
//
#include <hip/hip_runtime.h>

typedef __attribute__((ext_vector_type(2))) float v2f;
typedef __attribute__((ext_vector_type(8))) float v8f;

#define F32_PI_2 1.57079632679489662f  // rounds to float(np.pi/2)

// ---------------- FK: 9 transforms, tiny serial work ----------------
__device__ __forceinline__ void mm4(const float* A, const float* B, float* C) {
#pragma unroll
  for (int r = 0; r < 4; ++r)
#pragma unroll
    for (int c = 0; c < 4; ++c) {
      float s = 0.f;
#pragma unroll
      for (int k = 0; k < 4; ++k) s += A[r * 4 + k] * B[k * 4 + c];
      C[r * 4 + c] = s;
    }
}

__device__ __forceinline__ void dhm(float al, float a, float th, float d, float* M) {
  float ca = cosf(al), sa = sinf(al);
  float ct = cosf(th), st = sinf(th);
  M[0] = ct;      M[1] = -st;     M[2] = 0.f;  M[3] = a;
  M[4] = st * ca; M[5] = ct * ca; M[6] = -sa;  M[7] = -d * sa;
  M[8] = st * sa; M[9] = ct * sa; M[10] = ca;  M[11] = d * ca;
  M[12] = 0.f;    M[13] = 0.f;    M[14] = 0.f; M[15] = 1.f;
}

__global__ void fk_kernel(const float* __restrict__ joint,
                          const float* __restrict__ baseT,
                          const float* __restrict__ linkD,
                          const float* __restrict__ toolD,
                          float* __restrict__ Ts /* 9*16 floats */) {
  if (threadIdx.x != 0 || blockIdx.x != 0) return;

  const float alpha[7] = {F32_PI_2, -F32_PI_2, F32_PI_2, 0.f, -F32_PI_2, -F32_PI_2, -F32_PI_2};
  const float aarr[7]  = {0.f, 0.f, 0.f, 0.f, 0.f, 0.091f, 0.f};
  const float d0[7]    = {0.f, 0.f, 0.f, 4.16f, 0.f, 0.f, 0.106f};
  const float off[7]   = {F32_PI_2, -F32_PI_2, -4.389f, 0.f, -F32_PI_2, -F32_PI_2, F32_PI_2};
  const bool  pris[7]  = {false, false, true, false, false, false, false};

  float T[16] = {1,0,0,0, 0,1,0,0, 0,0,1,0, 0,0,0,1};  // T_N_0
  float Tr[16], M[16], tmp[16];
  for (int k = 0; k < 16; ++k) Tr[k] = baseT[k];       // baseT @ I

  for (int i = 0; i < 7; ++i) {
    for (int k = 0; k < 16; ++k) Ts[i * 16 + k] = Tr[k];
    float ji = (i < 6) ? joint[i] : 0.f;
    if (pris[i]) dhm(alpha[i], aarr[i], 0.f, ji + off[i], M);
    else         dhm(alpha[i], aarr[i], ji + off[i], d0[i], M);
    mm4(T, M, tmp);
    for (int k = 0; k < 16; ++k) T[k] = tmp[k];        // T_N_0 @= M
    mm4(T, linkD + i * 16, tmp);                       // T_N_0 @ linkD[i]
    mm4(baseT, tmp, Tr);                               // baseT @ (...)
  }

  float ta = joint[6] * F32_PI_2;
  float bT[16];
  mm4(baseT, T, bT);                                   // baseT @ T_N_0
  for (int t = 0; t < 2; ++t) {
    float ang = (t == 0) ? -ta : ta;
    dhm(0.f, 0.f, ang, 0.f, M);
    mm4(bT, toolD + t * 16, tmp);                      // (baseT@T) @ toolD[t]
    mm4(tmp, M, Tr);                                   // ... @ Mt
    for (int k = 0; k < 16; ++k) Ts[(7 + t) * 16 + k] = Tr[k];
  }
}

// -------- Vertex transform: V_WMMA_F32_16X16X4_F32, 16 verts / wave --------
// A (16x4): rows 0..2 = [R | tr] of transform m, rows 3..15 = 0
// B (4x16): column n = (x_n, y_n, z_n, 1) for 16 vertices
// D (16x16): lane n<16 holds (x',y',z') of vertex n in VGPRs 0..2
__global__ void verts_wmma_kernel(const float* __restrict__ vin,
                                  const float* __restrict__ Ts,
                                  float* __restrict__ vout,
                                  int V, int groupsPerMesh, int totalGroups) {
  int wavesPerBlock = blockDim.x >> 5;
  int g = blockIdx.x * wavesPerBlock + ((int)threadIdx.x >> 5);
  if (g >= totalGroups) return;                 // wave-uniform exit
  int lane = threadIdx.x & 31;
  int m = g / groupsPerMesh;
  int v0 = (g - m * groupsPerMesh) * 16;

  // A operand
  const float* T = Ts + m * 16;
  int mrow = lane & 15;
  v2f a; a.x = 0.f; a.y = 0.f;
  if (mrow < 3) {
    int kb = (lane < 16) ? 0 : 2;
    a.x = T[mrow * 4 + kb];
    a.y = T[mrow * 4 + kb + 1];
  }

  // B operand
  const float* vp = vin + ((size_t)m * V + v0) * 3;
  int vidx = lane & 15;
  bool inb = (v0 + vidx) < V;
  v2f b; b.x = 0.f; b.y = 1.f;
  if (lane < 16) {
    if (inb) { b.x = vp[vidx * 3 + 0]; b.y = vp[vidx * 3 + 1]; }
    else     { b.y = 0.f; }
  } else {
    if (inb) { b.x = vp[vidx * 3 + 2]; }
  }

  v8f c = {};
  c = __builtin_amdgcn_wmma_f32_16x16x4_f32(
      /*neg_a=*/false, a, /*neg_b=*/false, b,
      /*c_mod=*/(short)0, c, /*reuse_a=*/false, /*reuse_b=*/false);

  if (lane < 16 && inb) {
    float* o = vout + ((size_t)m * V + v0 + lane) * 3;
    o[0] = c[0];
    o[1] = c[1];
    o[2] = c[2];
  }
}

// ---------------- Faces: face_ids + m*V ----------------
// 2D grid: blockIdx.y = mesh m (no per-element integer divide),
// float4 vectorized over the F*3 elements of each mesh, scalar tail.
__global__ void faces_kernel(const float* __restrict__ fin,
                             float* __restrict__ fout,
                             int perMesh /* F*3 */, float Vf) {
  int m = blockIdx.y;
  float offs = (float)m * Vf;
  const float* in = fin + (size_t)m * perMesh;
  float* out = fout + (size_t)m * perMesh;

  int nvec = perMesh >> 2;  // # of float4
  int i = blockIdx.x * blockDim.x + threadIdx.x;
  if (i < nvec) {
    float4 v = ((const float4*)in)[i];
    v.x += offs; v.y += offs; v.z += offs; v.w += offs;
    ((float4*)out)[i] = v;
  }
  // tail (perMesh % 4 elements), handled by the first few threads of block 0
  int tail = perMesh & 3;
  if (blockIdx.x == 0 && (int)threadIdx.x < tail) {
    int t = (nvec << 2) + threadIdx.x;
    out[t] = in[t] + offs;
  }
}

extern "C" void kernel_launch(void* const* d_in, const int* in_sizes, int n_in,
                              void* d_out, int out_size, void* d_ws, size_t ws_size,
                              hipStream_t stream) {
  const float* joint = (const float*)d_in[0];
  const float* baseT = (const float*)d_in[1];
  const float* linkD = (const float*)d_in[2];
  const float* toolD = (const float*)d_in[3];
  const float* verts = (const float*)d_in[4];
  const float* faces = (const float*)d_in[5];
  const float* atlas = (const float*)d_in[6];
  float* out = (float*)d_out;

  const int NM = 9;
  int V = in_sizes[4] / (NM * 3);
  int F = in_sizes[5] / (NM * 3);
  float* Ts = (float*)d_ws;  // 9*16 floats of scratch

  // 1) forward kinematics (tiny)
  fk_kernel<<<1, 32, 0, stream>>>(joint, baseT, linkD, toolD, Ts);

  // 2) vertex transform via f32 WMMA: 16 vertices per wave
  int gpm = (V + 15) / 16;
  int totalGroups = NM * gpm;
  const int wavesPerBlock = 8;  // 256 threads
  int vblocks = (totalGroups + wavesPerBlock - 1) / wavesPerBlock;
  verts_wmma_kernel<<<vblocks, wavesPerBlock * 32, 0, stream>>>(
      verts, Ts, out, V, gpm, totalGroups);

  // 3) faces offset add: grid.y = mesh index, float4 over F*3
  int perMesh = F * 3;
  int nvec = perMesh >> 2;
  dim3 fgrid((nvec + 255) / 256, NM, 1);
  float* faces_out = out + (size_t)NM * V * 3;
  faces_kernel<<<fgrid, 256, 0, stream>>>(faces, faces_out, perMesh, (float)V);

  // 4) atlas: pure reshape -> flat device-to-device copy (bandwidth-optimal)
  float* atlas_out = out + (size_t)NM * V * 3 + (size_t)NM * F * 3;
  hipMemcpyAsync(atlas_out, atlas, (size_t)in_sizes[6] * sizeof(float),
                 hipMemcpyDeviceToDevice, stream);
}